// captioner_4793183502895
// MI455X (gfx1250) — compile-verified
//
#include <hip/hip_runtime.h>
#include <math.h>

typedef __attribute__((ext_vector_type(2))) float v2f;
typedef __attribute__((ext_vector_type(8))) float v8f;

#define IN_W   2048
#define C1_W   2046
#define S1     (2046*2046)
#define P1_W   1023
#define SP1    (1023*1023)
#define C2_W   1021
#define S2     (1021*1021)
#define P2_W   510
#define SP2    (510*510)

#define TILES_PER_WAVE 4

// workspace offsets (in floats); conv2 output overlays conv1 output (dead by then)
#define OFF_CONV1 0
#define OFF_CONV2 0
#define OFF_POOL1 (4*S1)
#define OFF_POOL2 (OFF_POOL1 + 4*SP1)
#define OFF_STAT1 (OFF_POOL2 + 16*SP2)   // sum[4],sumsq[4],mean[4],inv[4]
#define OFF_STAT2 (OFF_STAT1 + 16)       // sum[16],sumsq[16],mean[16],inv[16]
#define OFF_VLAD  (OFF_STAT2 + 64)       // vlad acc [12*16] + asum[12]
#define OFF_Z     (OFF_VLAD + 204)       // z[100]

__global__ void k_init(float* p, int n) {
    for (int i = blockIdx.x*blockDim.x + threadIdx.x; i < n; i += blockDim.x*gridDim.x)
        p[i] = 0.f;
}

// conv1 3->4, 3x3 VALID, + per-channel sum/sumsq for training-mode BN
__global__ void k_conv1(const float* __restrict__ x, const float* __restrict__ w,
                        const float* __restrict__ b, float* __restrict__ out,
                        float* __restrict__ stats) {
    __shared__ float ssum[4], ssq[4];
    if (threadIdx.x < 4) { ssum[threadIdx.x] = 0.f; ssq[threadIdx.x] = 0.f; }
    __syncthreads();
    int idx = blockIdx.x*blockDim.x + threadIdx.x;
    if (idx < S1) {
        int yy = idx / C1_W, xx = idx % C1_W;
        float v[27];
        #pragma unroll
        for (int i = 0; i < 3; i++)
            #pragma unroll
            for (int ky = 0; ky < 3; ky++)
                #pragma unroll
                for (int kx = 0; kx < 3; kx++)
                    v[i*9 + ky*3 + kx] = x[i*(IN_W*IN_W) + (yy+ky)*IN_W + (xx+kx)];
        #pragma unroll
        for (int c = 0; c < 4; c++) {
            float acc = b[c];
            #pragma unroll
            for (int t = 0; t < 27; t++) acc += v[t]*w[c*27 + t];
            out[c*S1 + idx] = acc;
            atomicAdd(&ssum[c], acc);
            atomicAdd(&ssq[c], acc*acc);
        }
    }
    __syncthreads();
    if (threadIdx.x < 4) {
        atomicAdd(&stats[threadIdx.x],     ssum[threadIdx.x]);
        atomicAdd(&stats[4 + threadIdx.x], ssq[threadIdx.x]);
    }
}

__global__ void k_bnfin(float* stats, int C, float invN) {
    int c = threadIdx.x;
    if (c < C) {
        float m   = stats[c]*invN;
        float var = stats[C + c]*invN - m*m;
        stats[2*C + c] = m;
        stats[3*C + c] = rsqrtf(var + 1e-5f);
    }
}

// BN(train)+ReLU+maxpool2x2 stride2 VALID; in [C,inW,inW] -> out [C,outW,outW]
__global__ void k_bnpool(const float* __restrict__ in, float* __restrict__ out,
                         const float* __restrict__ stats, const float* __restrict__ g,
                         const float* __restrict__ bt, int C, int inW, int outW) {
    int idx = blockIdx.x*blockDim.x + threadIdx.x;
    int total = C*outW*outW;
    if (idx >= total) return;
    int c  = idx / (outW*outW);
    int r  = idx % (outW*outW);
    int oy = r / outW, ox = r % outW;
    float mean = stats[2*C + c], inv = stats[3*C + c], gg = g[c], bb = bt[c];
    const float* base = in + (size_t)c*inW*inW + (2*oy)*inW + 2*ox;
    float m = 0.f;  // post-ReLU values are >= 0
    #pragma unroll
    for (int dy = 0; dy < 2; dy++)
        #pragma unroll
        for (int dx = 0; dx < 2; dx++) {
            float v = base[dy*inW + dx];
            v = (v - mean)*inv*gg + bb;
            v = fmaxf(v, 0.f);
            m = fmaxf(m, v);
        }
    out[idx] = m;
}

// im2col flat offset for K-index kk (kk = 4 input channels x 3x3 taps, row-major)
__host__ __device__ constexpr int im2col_off(int kk) {
    return (kk/9)*SP1 + ((kk%9)/3)*P1_W + ((kk%9)%3);
}

// conv2 4->16, 3x3 VALID as implicit GEMM: D[16, S2] = W[16,36] @ im2col[36, S2]
// One wave -> TILES_PER_WAVE consecutive 16(ch) x 16(pixel) tiles;
// K=36 in 9 steps of V_WMMA_F32_16X16X4_F32. A fragments reused across tiles.
__global__ void k_conv2_wmma(const float* __restrict__ pooled1, const float* __restrict__ w2,
                             const float* __restrict__ b2, float* __restrict__ out,
                             float* __restrict__ stats, int numTiles) {
    __shared__ float ssum[16], ssq[16];
    if (threadIdx.x < 16) { ssum[threadIdx.x] = 0.f; ssq[threadIdx.x] = 0.f; }
    __syncthreads();
    int wave = threadIdx.x >> 5;
    int lane = threadIdx.x & 31;
    int hi   = lane >> 4;            // half-wave select (K offset +2 per ISA layout)
    int lo   = lane & 15;            // A: row m ; B: column n

    // A fragments: lane holds W[m=lo, kk] with kk = 4t + 2*hi + {0,1}
    v2f A[9];
    // im2col offsets: compile-time constants selected by hi (single v_cndmask each)
    int offx[9], offy[9];
    #pragma unroll
    for (int t = 0; t < 9; t++) {
        int kk = 4*t + 2*hi;
        A[t].x = w2[lo*36 + kk];
        A[t].y = w2[lo*36 + kk + 1];
        offx[t] = hi ? im2col_off(4*t + 2) : im2col_off(4*t);
        offy[t] = hi ? im2col_off(4*t + 3) : im2col_off(4*t + 1);
    }

    int tile0 = (blockIdx.x*(blockDim.x >> 5) + wave)*TILES_PER_WAVE;
    for (int u = 0; u < TILES_PER_WAVE; u++) {
        int tile = tile0 + u;
        if (tile >= numTiles) break;          // wave-uniform: EXEC all-1s at WMMA
        int n  = tile*16 + lo;
        int nc = (n < S2) ? n : (S2 - 1);     // clamp tail-tile addresses
        float msk = (n < S2) ? 1.f : 0.f;     // zero tail-tile B columns
        int yy = nc / C2_W, xx = nc % C2_W;
        const float* bp = pooled1 + yy*P1_W + xx;
        v8f acc = {0.f,0.f,0.f,0.f,0.f,0.f,0.f,0.f};
        #pragma unroll
        for (int t = 0; t < 9; t++) {
            v2f B;
            B.x = bp[offx[t]] * msk;
            B.y = bp[offy[t]] * msk;
            acc = __builtin_amdgcn_wmma_f32_16x16x4_f32(false, A[t], false, B,
                                                        (short)0, acc, false, false);
        }
        if (n < S2) {
            // C layout: VGPR j -> channel j + 8*hi, column n
            #pragma unroll
            for (int j = 0; j < 8; j++) {
                int mo = j + 8*hi;
                float val = acc[j] + b2[mo];
                out[(size_t)mo*S2 + n] = val;
                atomicAdd(&ssum[mo], val);
                atomicAdd(&ssq[mo],  val*val);
            }
        }
    }
    __syncthreads();
    if (threadIdx.x < 16) {
        atomicAdd(&stats[threadIdx.x],      ssum[threadIdx.x]);
        atomicAdd(&stats[16 + threadIdx.x], ssq[threadIdx.x]);
    }
}

// fused softmax(logits over 12 clusters) + VLAD accumulation
__global__ void k_vlad(const float* __restrict__ xf, const float* __restrict__ aggw,
                       const float* __restrict__ aggb, float* __restrict__ vacc) {
    __shared__ float sv[204];   // [12*16] vlad partial + [12] a-sum
    for (int i = threadIdx.x; i < 204; i += blockDim.x) sv[i] = 0.f;
    __syncthreads();
    int s = blockIdx.x*blockDim.x + threadIdx.x;
    if (s < SP2) {
        float xv[16];
        #pragma unroll
        for (int c = 0; c < 16; c++) xv[c] = xf[(size_t)c*SP2 + s];
        float lg[12]; float mx = -1e30f;
        #pragma unroll
        for (int k = 0; k < 12; k++) {
            float a = aggb[k];
            #pragma unroll
            for (int c = 0; c < 16; c++) a += aggw[k*16 + c]*xv[c];
            lg[k] = a; mx = fmaxf(mx, a);
        }
        float den = 0.f;
        #pragma unroll
        for (int k = 0; k < 12; k++) { lg[k] = expf(lg[k] - mx); den += lg[k]; }
        float r = 1.f/den;
        #pragma unroll
        for (int k = 0; k < 12; k++) {
            float a = lg[k]*r;
            atomicAdd(&sv[192 + k], a);
            #pragma unroll
            for (int c = 0; c < 16; c++) atomicAdd(&sv[k*16 + c], a*xv[c]);
        }
    }
    __syncthreads();
    for (int i = threadIdx.x; i < 204; i += blockDim.x) atomicAdd(&vacc[i], sv[i]);
}

// vlad -= asum*centres ; row L2-normalize ; z = fc_w @ vlad + fc_b
__global__ void k_vlad_fc(const float* __restrict__ vacc, const float* __restrict__ aggw,
                          const float* __restrict__ fcw, const float* __restrict__ fcb,
                          float* __restrict__ z) {
    __shared__ float vl[192];
    __shared__ float sn[12];
    int tid = threadIdx.x;
    if (tid < 192) {
        int k = tid >> 4;
        vl[tid] = vacc[tid] - vacc[192 + k]*(aggw[tid]*5.0f);   // centres = aggw/0.2
    }
    __syncthreads();
    if (tid < 12) {
        float s = 0.f;
        for (int c = 0; c < 16; c++) { float v = vl[tid*16 + c]; s += v*v; }
        sn[tid] = 1.f / fmaxf(sqrtf(s), 1e-12f);
    }
    __syncthreads();
    if (tid < 192) vl[tid] *= sn[tid >> 4];
    __syncthreads();
    if (tid < 100) {
        float a = fcb[tid];
        for (int q = 0; q < 192; q++) a += fcw[tid*192 + q]*vl[q];
        z[tid] = a;
    }
}

// sequential LSTM scan (pre-tanh output fed back as next input)
__global__ void k_lstm(const float* __restrict__ z, const float* __restrict__ wih,
                       const float* __restrict__ whh, const float* __restrict__ bih,
                       const float* __restrict__ bhh, const float* __restrict__ outw,
                       const float* __restrict__ outb, const float* __restrict__ h0,
                       const float* __restrict__ c0, const int* __restrict__ oplen,
                       float* __restrict__ dout) {
    __shared__ float h[50], c[50], inp[100], gates[200];
    int tid = threadIdx.x;
    if (tid < 50)  { h[tid] = h0[tid]; c[tid] = c0[tid]; }
    if (tid < 100) inp[tid] = z[tid];
    __syncthreads();
    int T = oplen[0];
    for (int t = 0; t < T; t++) {
        if (tid < 200) {
            float g = bih[tid] + bhh[tid];
            for (int i = 0; i < 100; i++) g += inp[i]*wih[tid*100 + i];
            for (int i = 0; i < 50;  i++) g += h[i]*whh[tid*50 + i];
            gates[tid] = g;
        }
        __syncthreads();
        if (tid < 50) {                               // PyTorch order: i,f,g,o
            float ig = 1.f/(1.f + expf(-gates[tid]));
            float fg = 1.f/(1.f + expf(-gates[50 + tid]));
            float gg = tanhf(gates[100 + tid]);
            float og = 1.f/(1.f + expf(-gates[150 + tid]));
            float cn = fg*c[tid] + ig*gg;
            c[tid] = cn;
            h[tid] = og*tanhf(cn);
        }
        __syncthreads();
        if (tid < 100) {
            float o = outb[tid];
            for (int i = 0; i < 50; i++) o += h[i]*outw[tid*50 + i];
            inp[tid] = o;                              // pre-tanh fed back
            dout[t*100 + tid] = tanhf(o);
        }
        __syncthreads();
    }
}

extern "C" void kernel_launch(void* const* d_in, const int* in_sizes, int n_in,
                              void* d_out, int out_size, void* d_ws, size_t ws_size,
                              hipStream_t stream) {
    (void)in_sizes; (void)n_in; (void)out_size; (void)ws_size;
    const float* x    = (const float*)d_in[0];
    const float* c1w  = (const float*)d_in[1];
    const float* c1b  = (const float*)d_in[2];
    const float* bn1g = (const float*)d_in[3];
    const float* bn1b = (const float*)d_in[4];
    const float* c2w  = (const float*)d_in[5];
    const float* c2b  = (const float*)d_in[6];
    const float* bn2g = (const float*)d_in[7];
    const float* bn2b = (const float*)d_in[8];
    const float* aggw = (const float*)d_in[9];
    const float* aggb = (const float*)d_in[10];
    const float* fcw  = (const float*)d_in[11];
    const float* fcb  = (const float*)d_in[12];
    const float* wih  = (const float*)d_in[13];
    const float* whh  = (const float*)d_in[14];
    const float* bih  = (const float*)d_in[15];
    const float* bhh  = (const float*)d_in[16];
    const float* outw = (const float*)d_in[17];
    const float* outb = (const float*)d_in[18];
    const float* h0   = (const float*)d_in[19];
    const float* c0   = (const float*)d_in[20];
    const int*   opl  = (const int*)d_in[21];

    float* ws    = (float*)d_ws;
    float* conv1 = ws + OFF_CONV1;
    float* conv2 = ws + OFF_CONV2;   // overlays conv1 (dead after pool1)
    float* pool1 = ws + OFF_POOL1;
    float* pool2 = ws + OFF_POOL2;
    float* st1   = ws + OFF_STAT1;
    float* st2   = ws + OFF_STAT2;
    float* vacc  = ws + OFF_VLAD;
    float* z     = ws + OFF_Z;
    float* dout  = (float*)d_out;

    k_init<<<1, 256, 0, stream>>>(st1, 16 + 64 + 204 + 100);
    k_conv1<<<(S1 + 255)/256, 256, 0, stream>>>(x, c1w, c1b, conv1, st1);
    k_bnfin<<<1, 32, 0, stream>>>(st1, 4, 1.f/(float)S1);
    k_bnpool<<<(4*SP1 + 255)/256, 256, 0, stream>>>(conv1, pool1, st1, bn1g, bn1b, 4, C1_W, P1_W);
    int nt = (S2 + 15)/16;
    int wavesNeeded = (nt + TILES_PER_WAVE - 1)/TILES_PER_WAVE;
    k_conv2_wmma<<<(wavesNeeded + 3)/4, 128, 0, stream>>>(pool1, c2w, c2b, conv2, st2, nt);
    k_bnfin<<<1, 32, 0, stream>>>(st2, 16, 1.f/(float)S2);
    k_bnpool<<<(16*SP2 + 255)/256, 256, 0, stream>>>(conv2, pool2, st2, bn2g, bn2b, 16, C2_W, P2_W);
    k_vlad<<<(SP2 + 255)/256, 256, 0, stream>>>(pool2, aggw, aggb, vacc);
    k_vlad_fc<<<1, 256, 0, stream>>>(vacc, aggw, fcw, fcb, z);
    k_lstm<<<1, 256, 0, stream>>>(z, wih, whh, bih, bhh, outw, outb, h0, c0, opl, dout);
}